// Post_Prob_GS_19971597927079
// MI455X (gfx1250) — compile-verified
//
#include <hip/hip_runtime.h>
#include <hip/hip_bf16.h>

typedef __attribute__((ext_vector_type(2))) float v2f;
typedef __attribute__((ext_vector_type(8))) float v8f;

#define HW   16384   // (512/4)*(512/4)
#define NPTS 256
#define NB   8

// ---------------------------------------------------------------------------
// Prep: per (b,n) build rank-6 feature row f[8] (K padded to 8) and half-logdet.
//   like(n,p) = f . [gx^2, gx*gy, gy^2, gx, gy, 1, 0, 0]   (coords centered @256)
// Slot 6 holds 0.5*log(det) (pairs with a zero B row -> inert in the GEMM).
// ---------------------------------------------------------------------------
__global__ __launch_bounds__(256) void pp_prep(const float* __restrict__ scale,
                                               const float* __restrict__ rot,
                                               const float* __restrict__ pts,
                                               float* __restrict__ F,
                                               float* __restrict__ H) {
  int t = blockIdx.x * blockDim.x + threadIdx.x;
  if (t >= NB * NPTS) return;
  float s0 = fminf(fmaxf(32.f * scale[2 * t + 0], 2.f), 100.f);
  float s1 = fminf(fmaxf(32.f * scale[2 * t + 1], 2.f), 100.f);
  float sx2 = s0 * s0, sy2 = s1 * s1;
  float det = sx2 * sy2;
  float ld  = logf(det);
  float th  = rot[t];
  float c = cosf(th), si = sinf(th);
  float cc = c * c, ss = si * si, cs = c * si;
  float cov_a = cc * sx2 + ss * sy2;
  float cov_b = cs * (sx2 - sy2);
  float cov_d = ss * sx2 + cc * sy2;
  float inv_a = cov_d / det;
  float inv_b = -(cov_b / det);
  float inv_d = cov_a / det;
  float px = pts[2 * t + 0] - 256.f;
  float py = pts[2 * t + 1] - 256.f;
  float* o = F + t * 8;
  o[0] = -0.5f * inv_a;
  o[1] = -inv_b;
  o[2] = -0.5f * inv_d;
  o[3] = inv_a * px + inv_b * py;
  o[4] = inv_b * px + inv_d * py;
  o[5] = -0.5f * (inv_a * px * px + 2.f * inv_b * px * py + inv_d * py * py) - 0.5f * ld;
  o[6] = 0.5f * ld;   // half-logdet in padding slot (B row 6 == 0)
  o[7] = 0.f;
  H[t] = 0.5f * ld;
}

// ---------------------------------------------------------------------------
// Main: one wave per (batch, 16-pixel tile). 16 WMMA M-tiles (all 256 points)
// kept in 128 accumulator VGPRs; softmax over 257 channels done in-register.
// A layout (f32 16x4): lane m=lid&15; VGPR0=K(2*half), VGPR1=K(2*half+1).
// B layout (f32 4x16): lane n=lid&15; VGPR0=K(2*half), VGPR1=K(2*half+1).
// C layout: VGPR r, lane l -> M = r + 8*(l>=16), N = l&15.
// ---------------------------------------------------------------------------
__global__ __launch_bounds__(256) void pp_main(const float* __restrict__ F,
                                               const float* __restrict__ H,
                                               float* __restrict__ out) {
  const int lid  = threadIdx.x & 31;
  const int wid  = threadIdx.x >> 5;
  const int half = lid >> 4;
  const int nc   = lid & 15;
  const int idx  = blockIdx.x * 8 + wid;   // 0..8191
  const int b    = idx >> 10;
  const int tile = idx & 1023;
  const int pbase = tile << 4;             // flat pixel base (row-contiguous)
  const int irow  = pbase >> 7;            // p / 128
  const int jcol  = pbase & 127;           // p % 128

  // centered grid coords for this lane's pixel column
  const float gx  = (float)(4 * (jcol + nc) + 2 - 256);
  const float gy  = (float)(4 * irow + 2 - 256);
  const float gx2 = gx * gx, gy2 = gy * gy, gxy = gx * gy;

  // B operands: kc0 rows K0..3 = [gx^2, gx*gy | gy^2, gx]
  //             kc1 rows K4..7 = [gy, 1 | 0, 0]
  v2f b0, b1;
  b0.x = half ? gy2 : gx2;
  b0.y = half ? gx  : gxy;
  b1.x = half ? 0.f : gy;
  b1.y = half ? 0.f : 1.f;

  const float* Fb = F + b * (NPTS * 8);
  const float* Hb = H + b * NPTS;

  v8f acc[16];
#pragma unroll
  for (int mt = 0; mt < 16; ++mt) {
    const float* fr = Fb + (mt * 16 + nc) * 8 + 2 * half;
    v2f a0, a1;
    a0.x = fr[0]; a0.y = fr[1];   // K = 2h, 2h+1
    a1.x = fr[4]; a1.y = fr[5];   // K = 4+2h, 5+2h  (slots 6,7 inert)
    v8f c = {};
    c = __builtin_amdgcn_wmma_f32_16x16x4_f32(false, a0, false, b0, (short)0, c, false, false);
    c = __builtin_amdgcn_wmma_f32_16x16x4_f32(false, a1, false, b1, (short)0, c, false, false);
    acc[mt] = c;
  }

  // Track max(like), and max(z)+its half-logdet (z = like + hld) per pixel.
  float mz = -3.0e38f, mh = 0.f, ml = -3.0e38f;
#pragma unroll
  for (int mt = 0; mt < 16; ++mt) {
#pragma unroll
    for (int r = 0; r < 8; ++r) {
      float h0 = Hb[mt * 16 + r];       // uniform -> scalar loads
      float h1 = Hb[mt * 16 + r + 8];
      float h  = half ? h1 : h0;
      float like = acc[mt][r];
      ml = fmaxf(ml, like);
      float z = like + h;
      if (z > mz) { mz = z; mh = h; }
    }
  }
  {  // combine lane halves (each half holds disjoint point rows)
    float omz = __shfl_xor(mz, 16, 32);
    float omh = __shfl_xor(mh, 16, 32);
    float oml = __shfl_xor(ml, 16, 32);
    if (omz > mz) { mz = omz; mh = omh; }
    ml = fmaxf(ml, oml);
  }

  const float d_const = -5000.f;               // -0.5*CUT_OFF^2
  const float logBG   = -1.8971199848858813f;  // log(0.15)
  float bk = d_const - mz - 2.f * mh + logBG;  // bk_z - log(det_sel/BG)
  float M  = fmaxf(ml, bk);

  float sum = 0.f;
#pragma unroll
  for (int mt = 0; mt < 16; ++mt) {
#pragma unroll
    for (int r = 0; r < 8; ++r) {
      float e = __expf(acc[mt][r] - M);
      acc[mt][r] = e;                  // keep exp in-register for the write pass
      sum += e;
    }
  }
  sum += __shfl_xor(sum, 16, 32);      // 256 point channels total
  float ebk = __expf(bk - M);
  sum += ebk;                          // +1 background channel
  float inv = 1.f / sum;

  // out layout (B, 257, HW): channel stride HW, pixels contiguous across lanes
  float* ob = out + (size_t)b * (size_t)(NPTS + 1) * HW + pbase + nc;
#pragma unroll
  for (int mt = 0; mt < 16; ++mt) {
#pragma unroll
    for (int r = 0; r < 8; ++r) {
      size_t n = (size_t)(mt * 16 + r + 8 * half);
      ob[n * (size_t)HW] = acc[mt][r] * inv;
    }
  }
  if (half == 0) ob[(size_t)NPTS * HW] = ebk * inv;
}

extern "C" void kernel_launch(void* const* d_in, const int* in_sizes, int n_in,
                              void* d_out, int out_size, void* d_ws, size_t ws_size,
                              hipStream_t stream) {
  const float* scale = (const float*)d_in[0];   // (8,256,2)
  const float* rot   = (const float*)d_in[1];   // (8,256,1)
  const float* pts   = (const float*)d_in[2];   // (8,256,2)
  // d_in[3] st_sizes: unused by the math (as in reference)
  (void)in_sizes; (void)n_in; (void)out_size; (void)ws_size;

  float* F = (float*)d_ws;            // 8*256*8 f32 = 64 KB
  float* H = F + NB * NPTS * 8;       // 8*256   f32 =  8 KB

  pp_prep<<<8, 256, 0, stream>>>(scale, rot, pts, F, H);
  pp_main<<<1024, 256, 0, stream>>>(F, H, (float*)d_out);
}